// LinearAttention_87153476370664
// MI455X (gfx1250) — compile-verified
//
#include <hip/hip_runtime.h>
#include <math.h>

// ---------------------------------------------------------------------------
// Linear attention block for MI455X (gfx1250), fp32 end-to-end via
// V_WMMA_F32_16X16X4_F32. Quadratic kv path factorized to S = k^T q (64x64).
// GEMMs: uniform-base + 32-bit lane offsets (saddr form), software-pipelined
// fragment loads so WMMAs overlap with in-flight global loads.
// ---------------------------------------------------------------------------

typedef __attribute__((ext_vector_type(2))) float v2f;
typedef __attribute__((ext_vector_type(8))) float v8f;

#define DIMX   1024
#define NTOK   1024
#define BATCH  2
#define NHEADS 16
#define HDIM   64
#define MROWS  (BATCH * NTOK)                 // 2048
#define QSZ    (BATCH * NHEADS * NTOK * HDIM) // 2097152 floats per activation buf
#define SCALEF 0.125f                         // 64^-0.5

__device__ __forceinline__ v8f wmma4(v2f a, v2f b, v8f c) {
  // D = A(16x4 f32) * B(4x16 f32) + C(16x16 f32)
  return __builtin_amdgcn_wmma_f32_16x16x4_f32(
      /*neg_a=*/false, a, /*neg_b=*/false, b,
      /*c_mod=*/(short)0, c, /*reuse_a=*/false, /*reuse_b=*/false);
}

__device__ __forceinline__ v2f ld2(const float* __restrict__ base, int off) {
  return *(const v2f*)(base + off);
}

__device__ __forceinline__ float softplus_f(float x) {
  return x > 0.0f ? x + log1pf(__expf(-x)) : log1pf(__expf(x));
}
__device__ __forceinline__ float gelu_f(float x) {
  return 0.5f * x * (1.0f + erff(x * 0.70710678118654752440f));
}

// ---------------------------------------------------------------------------
// K1: qkvg = x @ W^T + b, fused activations, scatter to per-head Q/K/V/G bufs
// grid (4096/256, 2048/64), block 256 (8 waves 2Mx4N; wave tile = 32M x 64N)
// ---------------------------------------------------------------------------
__global__ __launch_bounds__(256) void qkvg_gemm_kernel(
    const float* __restrict__ x, const float* __restrict__ w,
    const float* __restrict__ bias, float* __restrict__ qb,
    float* __restrict__ kb, float* __restrict__ vb, float* __restrict__ gb) {
  const int lane  = threadIdx.x & 31;
  const int l15   = lane & 15;
  const int khalf = (lane >> 4) << 1;  // 0 for lanes 0-15, 2 for lanes 16-31
  const int wave  = threadIdx.x >> 5;

  const int m0 = blockIdx.y * 64 + (wave >> 2) * 32;   // two 16-row A tiles
  const int j0 = blockIdx.x * 256 + (wave & 3) * 64;   // four 16-col B tiles

  // 32-bit per-lane element offsets; x/w stay uniform SGPR bases.
  const int ao0 = (m0 +      l15) * DIMX + khalf;
  const int ao1 = (m0 + 16 + l15) * DIMX + khalf;
  const int bo0 = (j0 +  0 + l15) * DIMX + khalf;
  const int bo1 = (j0 + 16 + l15) * DIMX + khalf;
  const int bo2 = (j0 + 32 + l15) * DIMX + khalf;
  const int bo3 = (j0 + 48 + l15) * DIMX + khalf;

  v8f c00 = {}, c10 = {}, c20 = {}, c30 = {};
  v8f c01 = {}, c11 = {}, c21 = {}, c31 = {};

  // software pipeline: preload K-step 0
  v2f a0 = ld2(x, ao0), a1 = ld2(x, ao1);
  v2f f0 = ld2(w, bo0), f1 = ld2(w, bo1), f2 = ld2(w, bo2), f3 = ld2(w, bo3);

#pragma unroll 2
  for (int k0 = 0; k0 < DIMX - 4; k0 += 4) {
    const int kn = k0 + 4;
    v2f a0n = ld2(x, ao0 + kn), a1n = ld2(x, ao1 + kn);
    v2f f0n = ld2(w, bo0 + kn), f1n = ld2(w, bo1 + kn);
    v2f f2n = ld2(w, bo2 + kn), f3n = ld2(w, bo3 + kn);
    c00 = wmma4(a0, f0, c00);  c01 = wmma4(a1, f0, c01);
    c10 = wmma4(a0, f1, c10);  c11 = wmma4(a1, f1, c11);
    c20 = wmma4(a0, f2, c20);  c21 = wmma4(a1, f2, c21);
    c30 = wmma4(a0, f3, c30);  c31 = wmma4(a1, f3, c31);
    a0 = a0n; a1 = a1n; f0 = f0n; f1 = f1n; f2 = f2n; f3 = f3n;
  }
  c00 = wmma4(a0, f0, c00);  c01 = wmma4(a1, f0, c01);
  c10 = wmma4(a0, f1, c10);  c11 = wmma4(a1, f1, c11);
  c20 = wmma4(a0, f2, c20);  c21 = wmma4(a1, f2, c21);
  c30 = wmma4(a0, f3, c30);  c31 = wmma4(a1, f3, c31);

  v8f accs[4][2] = {{c00, c01}, {c10, c11}, {c20, c21}, {c30, c31}};
#pragma unroll
  for (int t = 0; t < 4; ++t) {
    const int col = j0 + t * 16 + l15;
    const int s = col >> 10;         // 0:q 1:k 2:v 3:g
    const int h = (col >> 6) & 15;
    const int e = col & 63;
    const float bi = bias[col];
    float* dst = (s == 0) ? qb : (s == 1) ? kb : (s == 2) ? vb : gb;
#pragma unroll
    for (int r = 0; r < 2; ++r) {
      const int rbase = m0 + r * 16 + ((lane >> 4) << 3);
#pragma unroll
      for (int i = 0; i < 8; ++i) {
        const int row = rbase + i;   // 0..2047 = b*1024+n
        const int b = row >> 10, n = row & 1023;
        float val = accs[t][r][i] + bi;
        val = (s < 2) ? softplus_f(val) : gelu_f(val);
        dst[((b * NHEADS + h) * NTOK + n) * HDIM + e] = val;
      }
    }
  }
}

// ---------------------------------------------------------------------------
// K2: z[b,h,e] = 1 / (SCALE * sum_c q[c,e]*ksum[c] + DIM), grid = 32 blocks
// ---------------------------------------------------------------------------
__global__ __launch_bounds__(256) void zvec_kernel(
    const float* __restrict__ qb, const float* __restrict__ kb,
    float* __restrict__ zb) {
  __shared__ float ksum[NTOK];
  __shared__ float zpart[256];
  const int tid = threadIdx.x;
  const float* kbh = kb + (size_t)blockIdx.x * NTOK * HDIM;
  const float* qbh = qb + (size_t)blockIdx.x * NTOK * HDIM;

#pragma unroll
  for (int cc = 0; cc < 4; ++cc) {
    const int c = tid * 4 + cc;
    const float4* p = (const float4*)(kbh + c * HDIM);
    float s = 0.0f;
#pragma unroll
    for (int i = 0; i < 16; ++i) {
      float4 f = p[i];
      s += f.x + f.y + f.z + f.w;
    }
    ksum[c] = s;
  }
  __syncthreads();

  const int e = tid & 63, seg = tid >> 6;
  float part = 0.0f;
  for (int c = seg * 256; c < seg * 256 + 256; ++c)
    part += qbh[c * HDIM + e] * ksum[c];
  zpart[tid] = part;
  __syncthreads();

  if (tid < 64) {
    float z = zpart[tid] + zpart[tid + 64] + zpart[tid + 128] + zpart[tid + 192];
    zb[blockIdx.x * 64 + tid] = 1.0f / (z * SCALEF + (float)DIMX);
  }
}

// ---------------------------------------------------------------------------
// K3: per (b,h): S = SCALE * k^T q (64x64, WMMA over 1024 tokens, into LDS),
// then out = (v @ S + v) * z * g, stored with the scrambled reshape into y.
// grid = 32 blocks, block = 256 threads (8 waves)
// ---------------------------------------------------------------------------
__global__ __launch_bounds__(256) void attn_kernel(
    const float* __restrict__ qb, const float* __restrict__ kb,
    const float* __restrict__ vb, const float* __restrict__ gb,
    const float* __restrict__ zb, float* __restrict__ yb) {
  __shared__ float S[64 * 64];
  __shared__ float zl[64];
  const int tid = threadIdx.x;
  const int lane = tid & 31, wave = tid >> 5;
  const int l15 = lane & 15;
  const int khalf = (lane >> 4) << 1;
  const int bI = blockIdx.x >> 4, hI = blockIdx.x & 15;

  const float* qbh = qb + (size_t)blockIdx.x * NTOK * HDIM;  // uniform bases
  const float* kbh = kb + (size_t)blockIdx.x * NTOK * HDIM;
  const float* vbh = vb + (size_t)blockIdx.x * NTOK * HDIM;
  const float* gbh = gb + (size_t)blockIdx.x * NTOK * HDIM;

  if (tid < 64) zl[tid] = zb[blockIdx.x * 64 + tid];

  // ---- Phase 1: S[d,e] = SCALE * sum_c k[c,d]*q[c,e]  (pipelined loads)
  const int te  = (wave & 3) * 16;   // shared e-tile for this wave's two tiles
  const int td0 = (wave >> 2) * 16;  // d-tiles: td0 and td0+32
  v8f s0 = {}, s1 = {};
  {
    int ko = khalf * HDIM;           // element offset of row kk
    v2f bq, a0, a1, bqn, a0n, a1n;
    bq.x = qbh[ko + te + l15];        bq.y = qbh[ko + HDIM + te + l15];
    a0.x = kbh[ko + td0 + l15];       a0.y = kbh[ko + HDIM + td0 + l15];
    a1.x = kbh[ko + td0 + 32 + l15];  a1.y = kbh[ko + HDIM + td0 + 32 + l15];
#pragma unroll 2
    for (int c0 = 0; c0 < NTOK - 4; c0 += 4) {
      const int kn = ko + 4 * HDIM;
      bqn.x = qbh[kn + te + l15];        bqn.y = qbh[kn + HDIM + te + l15];
      a0n.x = kbh[kn + td0 + l15];       a0n.y = kbh[kn + HDIM + td0 + l15];
      a1n.x = kbh[kn + td0 + 32 + l15];  a1n.y = kbh[kn + HDIM + td0 + 32 + l15];
      s0 = wmma4(a0, bq, s0);
      s1 = wmma4(a1, bq, s1);
      bq = bqn; a0 = a0n; a1 = a1n; ko = kn;
    }
    s0 = wmma4(a0, bq, s0);
    s1 = wmma4(a1, bq, s1);
  }
  {
    const int dd = (lane >> 4) << 3;
#pragma unroll
    for (int i = 0; i < 8; ++i) {
      S[(td0 + dd + i) * 64 + te + l15]      = s0[i] * SCALEF;
      S[(td0 + 32 + dd + i) * 64 + te + l15] = s1[i] * SCALEF;
    }
  }
  __syncthreads();

  // ---- Phase 2: out[m,e] = (v @ S + v) * z * g  (K = 64, B-fragments from LDS)
  for (int mt = 0; mt < 8; ++mt) {
    const int m0 = (wave * 8 + mt) * 16;
    v8f c0 = {}, c1 = {}, c2 = {}, c3 = {};
    const int vo = (m0 + l15) * HDIM + khalf;
#pragma unroll
    for (int d0 = 0; d0 < HDIM; d0 += 4) {
      const int kk = d0 + khalf;
      v2f a = ld2(vbh, vo + d0);
      v2f f0, f1, f2, f3;
      f0.x = S[kk * 64 +  0 + l15]; f0.y = S[(kk + 1) * 64 +  0 + l15];
      f1.x = S[kk * 64 + 16 + l15]; f1.y = S[(kk + 1) * 64 + 16 + l15];
      f2.x = S[kk * 64 + 32 + l15]; f2.y = S[(kk + 1) * 64 + 32 + l15];
      f3.x = S[kk * 64 + 48 + l15]; f3.y = S[(kk + 1) * 64 + 48 + l15];
      c0 = wmma4(a, f0, c0);
      c1 = wmma4(a, f1, c1);
      c2 = wmma4(a, f2, c2);
      c3 = wmma4(a, f3, c3);
    }
    v8f accs[4] = {c0, c1, c2, c3};
    const int rbase = m0 + ((lane >> 4) << 3);
#pragma unroll
    for (int t = 0; t < 4; ++t) {
      const int e = t * 16 + l15;
      const float ze = zl[e];
#pragma unroll
      for (int i = 0; i < 8; ++i) {
        const int m = rbase + i;
        const float vv = vbh[m * HDIM + e];
        const float gg = gbh[m * HDIM + e];
        const float val = (accs[t][i] + vv) * ze * gg;
        // raw reshape (B,H,N,hd)->(B,N,dim): row = h*64+m/16, col = (m%16)*64+e
        const int row = bI * NTOK + hI * 64 + (m >> 4);
        const int col = (m & 15) * 64 + e;
        yb[(size_t)row * DIMX + col] = val;
      }
    }
  }
}

// ---------------------------------------------------------------------------
// K4: out = y @ W_proj^T + b_proj   (2048 x 1024 x 1024)
// grid (1024/256, 2048/64), block 256
// ---------------------------------------------------------------------------
__global__ __launch_bounds__(256) void proj_gemm_kernel(
    const float* __restrict__ y, const float* __restrict__ w,
    const float* __restrict__ bias, float* __restrict__ out) {
  const int lane  = threadIdx.x & 31;
  const int l15   = lane & 15;
  const int khalf = (lane >> 4) << 1;
  const int wave  = threadIdx.x >> 5;

  const int m0 = blockIdx.y * 64 + (wave >> 2) * 32;
  const int j0 = blockIdx.x * 256 + (wave & 3) * 64;

  const int ao0 = (m0 +      l15) * DIMX + khalf;
  const int ao1 = (m0 + 16 + l15) * DIMX + khalf;
  const int bo0 = (j0 +  0 + l15) * DIMX + khalf;
  const int bo1 = (j0 + 16 + l15) * DIMX + khalf;
  const int bo2 = (j0 + 32 + l15) * DIMX + khalf;
  const int bo3 = (j0 + 48 + l15) * DIMX + khalf;

  v8f c00 = {}, c10 = {}, c20 = {}, c30 = {};
  v8f c01 = {}, c11 = {}, c21 = {}, c31 = {};

  v2f a0 = ld2(y, ao0), a1 = ld2(y, ao1);
  v2f f0 = ld2(w, bo0), f1 = ld2(w, bo1), f2 = ld2(w, bo2), f3 = ld2(w, bo3);

#pragma unroll 2
  for (int k0 = 0; k0 < DIMX - 4; k0 += 4) {
    const int kn = k0 + 4;
    v2f a0n = ld2(y, ao0 + kn), a1n = ld2(y, ao1 + kn);
    v2f f0n = ld2(w, bo0 + kn), f1n = ld2(w, bo1 + kn);
    v2f f2n = ld2(w, bo2 + kn), f3n = ld2(w, bo3 + kn);
    c00 = wmma4(a0, f0, c00);  c01 = wmma4(a1, f0, c01);
    c10 = wmma4(a0, f1, c10);  c11 = wmma4(a1, f1, c11);
    c20 = wmma4(a0, f2, c20);  c21 = wmma4(a1, f2, c21);
    c30 = wmma4(a0, f3, c30);  c31 = wmma4(a1, f3, c31);
    a0 = a0n; a1 = a1n; f0 = f0n; f1 = f1n; f2 = f2n; f3 = f3n;
  }
  c00 = wmma4(a0, f0, c00);  c01 = wmma4(a1, f0, c01);
  c10 = wmma4(a0, f1, c10);  c11 = wmma4(a1, f1, c11);
  c20 = wmma4(a0, f2, c20);  c21 = wmma4(a1, f2, c21);
  c30 = wmma4(a0, f3, c30);  c31 = wmma4(a1, f3, c31);

  v8f accs[4][2] = {{c00, c01}, {c10, c11}, {c20, c21}, {c30, c31}};
#pragma unroll
  for (int t = 0; t < 4; ++t) {
    const int col = j0 + t * 16 + l15;
    const float bi = bias[col];
#pragma unroll
    for (int r = 0; r < 2; ++r) {
      const int rbase = m0 + r * 16 + ((lane >> 4) << 3);
#pragma unroll
      for (int i = 0; i < 8; ++i) {
        out[(size_t)(rbase + i) * DIMX + col] = accs[t][r][i] + bi;
      }
    }
  }
}

// ---------------------------------------------------------------------------
extern "C" void kernel_launch(void* const* d_in, const int* in_sizes, int n_in,
                              void* d_out, int out_size, void* d_ws,
                              size_t ws_size, hipStream_t stream) {
  const float* x      = (const float*)d_in[0];
  const float* w_qkvg = (const float*)d_in[1];
  const float* b_qkvg = (const float*)d_in[2];
  const float* w_proj = (const float*)d_in[3];
  const float* b_proj = (const float*)d_in[4];
  float* out = (float*)d_out;

  float* ws = (float*)d_ws;
  float* qb = ws;                 // (B,H,N,hd) each
  float* kb = qb + QSZ;
  float* vb = kb + QSZ;
  float* gb = vb + QSZ;
  float* zb = gb + QSZ;           // (B,H,hd) = 2048 floats (padded to 4096)
  float* yb = zb + 4096;          // (B*N, DIM) = 2M floats

  qkvg_gemm_kernel<<<dim3(16, 32), 256, 0, stream>>>(x, w_qkvg, b_qkvg,
                                                     qb, kb, vb, gb);
  zvec_kernel<<<dim3(32), 256, 0, stream>>>(qb, kb, zb);
  attn_kernel<<<dim3(32), 256, 0, stream>>>(qb, kb, vb, gb, zb, yb);
  proj_gemm_kernel<<<dim3(4, 32), 256, 0, stream>>>(yb, w_proj, b_proj, out);
}